// CustomBRepEncoder_73796128079942
// MI455X (gfx1250) — compile-verified
//
#include <hip/hip_runtime.h>
#include <hip/hip_bf16.h>

typedef __attribute__((ext_vector_type(2))) float v2f;
typedef __attribute__((ext_vector_type(8))) float v8f;

#define WDIM 128
#define KDIM 256
#define LDS_STRIDE 260   // 260 % 64 == 4 -> 16 rows spread over 16 banks; row = 1040B (16B aligned)

__device__ __forceinline__ float lrelu(float x) {
    return x >= 0.0f ? x : 0.01f * x;
}

// Monotone order-preserving float->u32 map.  flip(x) > 0 for any real float,
// so 0u serves as the "empty segment" sentinel (reference: empty -> 0).
__device__ __forceinline__ unsigned int flip_f32(float x) {
    unsigned int b = __float_as_uint(x);
    unsigned int m = (b & 0x80000000u) ? 0xFFFFFFFFu : 0x80000000u;
    return b ^ m;
}

__device__ __forceinline__ float unflip_f32(unsigned int u) {
    if (u == 0u) return 0.0f;   // empty segment
    unsigned int m = (u & 0x80000000u) ? 0x80000000u : 0xFFFFFFFFu;
    return __uint_as_float(u ^ m);
}

// ---------------------------------------------------------------------------
// Input embedding: out[row,c] = lrelu( sum_k x[row,k] * Wt[k,c] + b[c] )
// K is tiny (3/12/14) -> plain VALU, one thread per output element.
// ---------------------------------------------------------------------------
__global__ void embed_kernel(const float* __restrict__ x,
                             const float* __restrict__ Wt,
                             const float* __restrict__ b,
                             float* __restrict__ out,
                             int nrows, int cin) {
    int row = blockIdx.x;
    int c   = threadIdx.x;
    if (row >= nrows) return;
    float acc = b[c];
    for (int k = 0; k < cin; ++k)
        acc += x[row * cin + k] * Wt[k * WDIM + c];
    out[(size_t)row * WDIM + c] = lrelu(acc);
}

__global__ void init_max_kernel(unsigned int* __restrict__ maxbuf, int n4) {
    int i = blockIdx.x * blockDim.x + threadIdx.x;   // index in uint4 units
    if (i < n4) ((uint4*)maxbuf)[i] = make_uint4(0u, 0u, 0u, 0u);
}

// ---------------------------------------------------------------------------
// Segment max of (x_dst[dst] - x_src[src]) into maxbuf[dst].
// One wave32 per edge, float4 per lane (32*4 = 128 channels).
// Atomic traffic stays in the 192MB L2 (maxbuf <= 100MB).
// ---------------------------------------------------------------------------
__global__ void seg_max_kernel(const float* __restrict__ xs,
                               const float* __restrict__ xd,
                               const int* __restrict__ src_idx,
                               const int* __restrict__ dst_idx,
                               unsigned int* __restrict__ maxbuf,
                               int nedges) {
    int e = blockIdx.x * 8 + (threadIdx.x >> 5);
    int lane = threadIdx.x & 31;
    if (e >= nedges) return;
    int s = src_idx[e];
    int d = dst_idx[e];
    const float4 a = *(const float4*)(xs + (size_t)s * WDIM + lane * 4);
    const float4 b = *(const float4*)(xd + (size_t)d * WDIM + lane * 4);
    unsigned int* mp = maxbuf + (size_t)d * WDIM + lane * 4;
    atomicMax(mp + 0, flip_f32(b.x - a.x));
    atomicMax(mp + 1, flip_f32(b.y - a.y));
    atomicMax(mp + 2, flip_f32(b.z - a.z));
    atomicMax(mp + 3, flip_f32(b.w - a.w));
}

// ---------------------------------------------------------------------------
// out = x_dst + lrelu( concat(x_dst, maxes) @ Wm + bm )
// [n_dst,256] x [256,128] GEMM via V_WMMA_F32_16X16X4_F32 (exact fp32).
// Block = 256 threads = 8 waves; block owns 16 rows, wave w owns cols
// [16w, 16w+16).  A-tile (16x256) staged in LDS via b128; 64 WMMA per wave.
// Residual x_dst[m][col] is re-read from the LDS tile (k<128 half), so the
// epilogue is straight-line LDS reads + coalesced stores on full tiles.
//
// f32 16x16x4 VGPR layouts (ISA 7.12.2):
//   A (16x4):  lane l (l<16): M=l, holds K=0 (v0), K=1 (v1);
//              lane l+16:     M=l, holds K=2 (v0), K=3 (v1)
//   B (4x16):  v0: K=0 @ lanes0-15 (N=lane), K=2 @ lanes16-31;
//              v1: K=1 / K=3
//   C/D:       vgpr r, lane l -> D[M = r + 8*(l>>4)][N = l&15]
// ---------------------------------------------------------------------------
__global__ __launch_bounds__(256) void conv_gemm_kernel(
        const float* __restrict__ xd,
        const unsigned int* __restrict__ maxbuf,
        const float* __restrict__ Wm,
        const float* __restrict__ bm,
        float* __restrict__ out,
        int n_dst) {
    __shared__ float lds[16 * LDS_STRIDE];
    const int tid = threadIdx.x;
    const int m0  = blockIdx.x * 16;

    // Stage h = concat(x_dst, maxes) rows m0..m0+15 into LDS.
    // 16*256 floats = 1024 float4; 256 threads -> 4 b128 loads + stores each.
    for (int idx = tid; idx < 16 * KDIM / 4; idx += 256) {
        int r  = idx >> 6;          // row 0..15 (64 float4 per row)
        int k4 = (idx & 63) * 4;    // k = 0,4,...,252
        int m  = m0 + r;
        if (m >= n_dst) m = n_dst - 1;
        float4 v;
        if (k4 < WDIM) {
            v = *(const float4*)(xd + (size_t)m * WDIM + k4);
        } else {
            uint4 u = *(const uint4*)(maxbuf + (size_t)m * WDIM + (k4 - WDIM));
            v = make_float4(unflip_f32(u.x), unflip_f32(u.y),
                            unflip_f32(u.z), unflip_f32(u.w));
        }
        *(float4*)(lds + r * LDS_STRIDE + k4) = v;
    }
    __syncthreads();

    const int wave = tid >> 5;
    const int lane = tid & 31;
    const int col0 = wave * 16;
    const int rrow = lane & 15;          // row (A) / N (B)
    const int hi   = lane >> 4;          // half-wave select
    const int khi  = hi * 2;             // K offset 0 or 2

    v8f acc = {};
    #pragma unroll 4
    for (int k0 = 0; k0 < KDIM; k0 += 4) {
        v2f a, b;
        a.x = lds[rrow * LDS_STRIDE + k0 + khi];
        a.y = lds[rrow * LDS_STRIDE + k0 + khi + 1];
        b.x = Wm[(size_t)(k0 + khi) * WDIM + col0 + rrow];
        b.y = Wm[(size_t)(k0 + khi + 1) * WDIM + col0 + rrow];
        acc = __builtin_amdgcn_wmma_f32_16x16x4_f32(
            /*neg_a=*/false, a, /*neg_b=*/false, b,
            /*c_mod=*/(short)0, acc, /*reuse_a=*/false, /*reuse_b=*/false);
    }

    const int   col  = col0 + rrow;
    const float bias = bm[col];
    if (m0 + 16 <= n_dst) {
        // Full tile: straight-line epilogue, residual from LDS.
        #pragma unroll
        for (int r = 0; r < 8; ++r) {
            int   m   = m0 + r + 8 * hi;
            float res = lds[(r + 8 * hi) * LDS_STRIDE + col];
            out[(size_t)m * WDIM + col] = res + lrelu(acc[r] + bias);
        }
    } else {
        #pragma unroll
        for (int r = 0; r < 8; ++r) {
            int m = m0 + r + 8 * hi;
            if (m < n_dst) {
                float res = lds[(r + 8 * hi) * LDS_STRIDE + col];
                out[(size_t)m * WDIM + col] = res + lrelu(acc[r] + bias);
            }
        }
    }
}

// ---------------------------------------------------------------------------
extern "C" void kernel_launch(void* const* d_in, const int* in_sizes, int n_in,
                              void* d_out, int out_size, void* d_ws, size_t ws_size,
                              hipStream_t stream) {
    const float* vertices = (const float*)d_in[0];
    const float* edges    = (const float*)d_in[1];
    const float* faces    = (const float*)d_in[2];
    const int* etv_v   = (const int*)d_in[3];
    const int* etv_e   = (const int*)d_in[4];
    const int* fte_e   = (const int*)d_in[5];
    const int* fte_f   = (const int*)d_in[6];
    const int* ftf_src = (const int*)d_in[7];
    const int* ftf_dst = (const int*)d_in[8];
    const float* Wv   = (const float*)d_in[9];
    const float* bv   = (const float*)d_in[10];
    const float* We   = (const float*)d_in[11];
    const float* be   = (const float*)d_in[12];
    const float* Wf   = (const float*)d_in[13];
    const float* bf   = (const float*)d_in[14];
    const float* Wv2e = (const float*)d_in[15];
    const float* bv2e = (const float*)d_in[16];
    const float* We2f = (const float*)d_in[17];
    const float* be2f = (const float*)d_in[18];
    const float* Wm0  = (const float*)d_in[19];
    const float* bm0  = (const float*)d_in[20];
    const float* Wm1  = (const float*)d_in[21];
    const float* bm1  = (const float*)d_in[22];

    const int NV   = in_sizes[0] / 3;
    const int NE   = in_sizes[1] / 12;
    const int NF   = in_sizes[2] / 14;
    const int N_EV = in_sizes[3];
    const int N_FE = in_sizes[5];
    const int N_FF = in_sizes[7];

    const size_t NVW  = (size_t)NV * WDIM;
    const size_t NEW_ = (size_t)NE * WDIM;
    const size_t NFW  = (size_t)NF * WDIM;

    float* ws  = (float*)d_ws;
    float* xv  = ws;                 // NV x 128
    float* xe  = xv + NVW;           // NE x 128
    float* xf  = xe + NEW_;          // NF x 128
    float* tmp = xf + NFW;           // NE x 128 capacity (ping-pong)
    unsigned int* maxbuf = (unsigned int*)(tmp + NEW_);  // NE x 128 capacity

    // Input embeddings.
    embed_kernel<<<NV, WDIM, 0, stream>>>(vertices, Wv, bv, xv, NV, 3);
    embed_kernel<<<NE, WDIM, 0, stream>>>(edges,    We, be, xe, NE, 12);
    embed_kernel<<<NF, WDIM, 0, stream>>>(faces,    Wf, bf, xf, NF, 14);

    auto conv = [&](const float* xs, const float* xdst,
                    const int* si, const int* di, int nedges, int ndst,
                    const float* Wm, const float* bmv, float* outbuf) {
        int n4 = ndst * WDIM / 4;    // WDIM divisible by 4
        init_max_kernel<<<(n4 + 255) / 256, 256, 0, stream>>>(maxbuf, n4);
        seg_max_kernel<<<(nedges + 7) / 8, 256, 0, stream>>>(xs, xdst, si, di,
                                                             maxbuf, nedges);
        conv_gemm_kernel<<<(ndst + 15) / 16, 256, 0, stream>>>(xdst, maxbuf, Wm,
                                                               bmv, outbuf, ndst);
    };

    // V2E:  x_e' = conv(x_v -> x_e)            result in tmp
    conv(xv,  xe,  etv_v,   etv_e,   N_EV, NE, Wv2e, bv2e, tmp);
    // E2F:  x_f' = conv(x_e' -> x_f)           result overwrites tmp (x_e' dead)
    conv(tmp, xf,  fte_e,   fte_f,   N_FE, NF, We2f, be2f, tmp);
    // FF0:  x_f'' = conv(x_f' -> x_f')         result in xf (old xf dead)
    conv(tmp, tmp, ftf_src, ftf_dst, N_FF, NF, Wm0,  bm0,  xf);
    // FF1:  final = conv(x_f'' -> x_f'')       straight into d_out
    conv(xf,  xf,  ftf_src, ftf_dst, N_FF, NF, Wm1,  bm1,  (float*)d_out);
}